// GAT_81020263072058
// MI455X (gfx1250) — compile-verified
//
#include <hip/hip_runtime.h>
#include <math.h>

// Problem constants (from reference)
#define NN      50000
#define FEAT    128
#define HH      2
#define CC      64
#define EB      800000
#define EE      (EB + NN)      // edges + self loops
#define GG      64
#define OUTC    10
#define NEG_SLOPE 0.2f

#define MTILES  (NN / 16)      // 3125 row tiles
#define WAVES_PER_BLOCK 8

typedef float v2f __attribute__((ext_vector_type(2)));
typedef float v8f __attribute__((ext_vector_type(8)));

static inline unsigned cdiv_u(long long a, long long b) { return (unsigned)((a + b - 1) / b); }

// ---------------------------------------------------------------------------
// Fold attention vectors into the weight matrix:
//   wsv[h][f] = sum_c Ws[f, h*C+c] * att_s[h,c]   (so a_s = X @ wsv[h])
// ---------------------------------------------------------------------------
__global__ void gat_attvec_kernel(const float* __restrict__ Ws, const float* __restrict__ Wd,
                                  const float* __restrict__ att_s, const float* __restrict__ att_d,
                                  float* __restrict__ wsv, float* __restrict__ wdv) {
    int t = blockIdx.x * blockDim.x + threadIdx.x;
    if (t >= HH * FEAT) return;
    int h = t / FEAT, f = t % FEAT;
    float s = 0.f, d = 0.f;
    for (int c = 0; c < CC; ++c) {
        s += Ws[f * (HH * CC) + h * CC + c] * att_s[h * CC + c];
        d += Wd[f * (HH * CC) + h * CC + c] * att_d[h * CC + c];
    }
    wsv[h * FEAT + f] = s;
    wdv[h * FEAT + f] = d;
}

// ---------------------------------------------------------------------------
// Out[N,128] = A[N,128] @ W[128,128] via V_WMMA_F32_16X16X4_F32.
// W is staged once per block into LDS (64 KB of the WGP's 320 KB); each wave
// then computes a full 16x128 row-block: 8 accumulators share one A fragment
// per K-step (A read exactly once from HBM, B served by ds_load).
// A layout (16x4 f32): lane&15 = row M; half-wave picks K pair {k0,k0+1} vs {k0+2,k0+3}.
// B layout (4x16 f32): lane&15 = col N; half-wave picks same K pair.
// C/D (16x16 f32, 8 VGPRs): vgpr i -> row i (lanes 0-15) / row i+8 (lanes 16-31).
// ---------------------------------------------------------------------------
__global__ void __launch_bounds__(WAVES_PER_BLOCK * 32)
gat_gemm_wmma_f32(const float* __restrict__ A, const float* __restrict__ W,
                  float* __restrict__ Out) {
    __shared__ float Wl[FEAT * FEAT];          // 64 KB
    // Cooperative W load (float4, coalesced): 16384 floats / 256 threads
    for (int i = threadIdx.x; i < (FEAT * FEAT) / 4; i += WAVES_PER_BLOCK * 32)
        ((float4*)Wl)[i] = ((const float4*)W)[i];
    __syncthreads();

    const int wave  = (int)threadIdx.x >> 5;
    const int tileM = (int)blockIdx.x * WAVES_PER_BLOCK + wave;
    if (tileM < MTILES) {                      // wave-uniform guard (EXEC all-ones inside)
        const int lane = (int)threadIdx.x & 31;
        const int half = lane >> 4;            // 0: lanes 0-15, 1: lanes 16-31
        const int l15  = lane & 15;
        const int row  = tileM * 16 + l15;

        v8f acc[8];
#pragma unroll
        for (int j = 0; j < 8; ++j) acc[j] = (v8f){0.f, 0.f, 0.f, 0.f, 0.f, 0.f, 0.f, 0.f};

#pragma unroll 2
        for (int k0 = 0; k0 < FEAT; k0 += 4) {
            const int kk = k0 + (half << 1);
            v2f a, b[8];
            a.x = A[row * FEAT + kk];
            a.y = A[row * FEAT + kk + 1];
            __builtin_prefetch(&A[row * FEAT + kk + 4], 0, 1);   // global_prefetch_b8
#pragma unroll
            for (int j = 0; j < 8; ++j) {
                b[j].x = Wl[kk * FEAT + j * 16 + l15];
                b[j].y = Wl[(kk + 1) * FEAT + j * 16 + l15];
            }
#pragma unroll
            for (int j = 0; j < 8; ++j) {
                acc[j] = __builtin_amdgcn_wmma_f32_16x16x4_f32(
                    /*neg_a=*/false, a, /*neg_b=*/false, b[j],
                    /*c_mod=*/(short)0, acc[j], /*reuse_a=*/false, /*reuse_b=*/false);
            }
        }

#pragma unroll
        for (int j = 0; j < 8; ++j)
#pragma unroll
            for (int i = 0; i < 8; ++i)
                Out[(tileM * 16 + i + half * 8) * FEAT + j * 16 + l15] = acc[j][i];
    }
}

// ---------------------------------------------------------------------------
// a_s[n,h] = dot(X[n,:], wsv[h,:]);  a_d[n,h] = dot(X[n,:], wdv[h,:])
// ---------------------------------------------------------------------------
__global__ void gat_attscore_kernel(const float* __restrict__ X,
                                    const float* __restrict__ wsv, const float* __restrict__ wdv,
                                    float* __restrict__ aS, float* __restrict__ aD) {
    int t = blockIdx.x * blockDim.x + threadIdx.x;
    if (t >= NN * HH) return;
    int node = t >> 1, h = t & 1;              // HH == 2
    float s = 0.f, d = 0.f;
    const float* xp = X + node * FEAT;
    const float* sp = wsv + h * FEAT;
    const float* dp = wdv + h * FEAT;
    for (int f = 0; f < FEAT; ++f) {
        float xv = xp[f];
        s += xv * sp[f];
        d += xv * dp[f];
    }
    aS[t] = s;
    aD[t] = d;
}

__global__ void gat_zero_kernel(float* __restrict__ p, int n) {
    int t = blockIdx.x * blockDim.x + threadIdx.x;
    if (t < n) p[t] = 0.f;
}

// Monotonic order-preserving float <-> uint encoding (for atomicMax on u32)
__device__ __forceinline__ unsigned enc_f32(float f) {
    unsigned u = __float_as_uint(f);
    return (u & 0x80000000u) ? ~u : (u | 0x80000000u);
}
__device__ __forceinline__ float dec_f32(unsigned u) {
    return (u & 0x80000000u) ? __uint_as_float(u ^ 0x80000000u) : __uint_as_float(~u);
}

__device__ __forceinline__ void edge_ends(const int* __restrict__ ei, int e, int& s, int& d) {
    if (e < EB) { s = ei[e]; d = ei[EB + e]; }
    else        { s = e - EB; d = e - EB; }   // self loop
}

// ---------------------------------------------------------------------------
// Pass 1: per-destination segment max of leaky_relu(a_s[src]+a_d[dst])
// ---------------------------------------------------------------------------
__global__ void gat_edgemax_kernel(const int* __restrict__ ei,
                                   const float* __restrict__ aS, const float* __restrict__ aD,
                                   unsigned* __restrict__ menc) {
    int t = blockIdx.x * blockDim.x + threadIdx.x;
    if (t >= EE * HH) return;
    int e = t >> 1, h = t & 1;
    int src, dst; edge_ends(ei, e, src, dst);
    float sc = aS[src * HH + h] + aD[dst * HH + h];
    sc = (sc > 0.f) ? sc : NEG_SLOPE * sc;
    atomicMax(&menc[dst * HH + h], enc_f32(sc));
}

// ---------------------------------------------------------------------------
// Pass 2: ew = exp(score - max[dst]); denom[dst] += ew
// ---------------------------------------------------------------------------
__global__ void gat_edgeexp_kernel(const int* __restrict__ ei,
                                   const float* __restrict__ aS, const float* __restrict__ aD,
                                   const unsigned* __restrict__ menc,
                                   float* __restrict__ denom, float* __restrict__ ew) {
    int t = blockIdx.x * blockDim.x + threadIdx.x;
    if (t >= EE * HH) return;
    int e = t >> 1, h = t & 1;
    int src, dst; edge_ends(ei, e, src, dst);
    float sc = aS[src * HH + h] + aD[dst * HH + h];
    sc = (sc > 0.f) ? sc : NEG_SLOPE * sc;
    float m = dec_f32(menc[dst * HH + h]);
    float w = expf(sc - m);
    ew[t] = w;
    atomicAdd(&denom[dst * HH + h], w);
}

// ---------------------------------------------------------------------------
// Pass 3: out[dst, c] += (ew/denom[dst]) * xs[src, c]
// One thread per 4 channels -> one wave per edge row; the xs[src] gather is a
// coalesced 512B row read (global_load_b128 per thread).
// ---------------------------------------------------------------------------
__global__ void gat_aggregate_kernel(const int* __restrict__ ei,
                                     const float* __restrict__ ew, const float* __restrict__ denom,
                                     const float* __restrict__ xs, float* __restrict__ outb) {
    int t = blockIdx.x * blockDim.x + threadIdx.x;
    if (t >= EE * (FEAT / 4)) return;
    int q  = t & 31;                           // 4-channel chunk within the row
    int e  = t >> 5;
    int c0 = q << 2;
    int h  = c0 >> 6;                          // CC == 64
    int src, dst; edge_ends(ei, e, src, dst);
    float alpha = ew[e * HH + h] / denom[dst * HH + h];
    float4 xv = *(const float4*)&xs[src * FEAT + c0];
    float* op = &outb[dst * FEAT + c0];
    atomicAdd(op + 0, alpha * xv.x);
    atomicAdd(op + 1, alpha * xv.y);
    atomicAdd(op + 2, alpha * xv.z);
    atomicAdd(op + 3, alpha * xv.w);
}

__global__ void gat_biasrelu_kernel(float* __restrict__ buf, const float* __restrict__ bias) {
    int t = blockIdx.x * blockDim.x + threadIdx.x;
    if (t >= NN * FEAT) return;
    float v = buf[t] + bias[t & (FEAT - 1)];
    buf[t] = v > 0.f ? v : 0.f;
}

// ---------------------------------------------------------------------------
// Mean pool per graph + final linear
// ---------------------------------------------------------------------------
__global__ void gat_pool_kernel(const float* __restrict__ hb, const int* __restrict__ batch,
                                float* __restrict__ sums, float* __restrict__ cnt) {
    int t = blockIdx.x * blockDim.x + threadIdx.x;
    if (t >= NN * FEAT) return;
    int n = t >> 7, j = t & (FEAT - 1);
    int g = batch[n];
    atomicAdd(&sums[g * FEAT + j], hb[t]);
    if (j == 0) atomicAdd(&cnt[g], 1.f);
}

__global__ void gat_final_kernel(const float* __restrict__ sums, const float* __restrict__ cnt,
                                 const float* __restrict__ Wl, const float* __restrict__ bl,
                                 float* __restrict__ out) {
    int t = blockIdx.x * blockDim.x + threadIdx.x;
    if (t >= GG * OUTC) return;
    int g = t / OUTC, o = t % OUTC;
    float inv = 1.f / fmaxf(cnt[g], 1.f);
    float acc = bl[o];
    for (int j = 0; j < FEAT; ++j)
        acc += (sums[g * FEAT + j] * inv) * Wl[j * OUTC + o];
    out[t] = acc;
}

// ---------------------------------------------------------------------------
// One GAT layer
// ---------------------------------------------------------------------------
static void run_gat_layer(const float* X, const float* Ws, const float* Wd,
                          const float* att_s, const float* att_d, const float* bias,
                          const int* ei,
                          float* xs, float* aS, float* aD, unsigned* menc,
                          float* denom, float* ew, float* wsv, float* wdv,
                          float* outbuf, hipStream_t stream) {
    gat_attvec_kernel<<<1, HH * FEAT, 0, stream>>>(Ws, Wd, att_s, att_d, wsv, wdv);
    gat_gemm_wmma_f32<<<cdiv_u(MTILES, WAVES_PER_BLOCK), WAVES_PER_BLOCK * 32, 0, stream>>>(X, Ws, xs);
    gat_attscore_kernel<<<cdiv_u((long long)NN * HH, 256), 256, 0, stream>>>(X, wsv, wdv, aS, aD);
    gat_zero_kernel<<<cdiv_u((long long)NN * HH, 256), 256, 0, stream>>>((float*)menc, NN * HH);
    gat_zero_kernel<<<cdiv_u((long long)NN * HH, 256), 256, 0, stream>>>(denom, NN * HH);
    gat_zero_kernel<<<cdiv_u((long long)NN * FEAT, 256), 256, 0, stream>>>(outbuf, NN * FEAT);
    gat_edgemax_kernel<<<cdiv_u((long long)EE * HH, 256), 256, 0, stream>>>(ei, aS, aD, menc);
    gat_edgeexp_kernel<<<cdiv_u((long long)EE * HH, 256), 256, 0, stream>>>(ei, aS, aD, menc, denom, ew);
    gat_aggregate_kernel<<<cdiv_u((long long)EE * (FEAT / 4), 256), 256, 0, stream>>>(ei, ew, denom, xs, outbuf);
    gat_biasrelu_kernel<<<cdiv_u((long long)NN * FEAT, 256), 256, 0, stream>>>(outbuf, bias);
}

extern "C" void kernel_launch(void* const* d_in, const int* in_sizes, int n_in,
                              void* d_out, int out_size, void* d_ws, size_t ws_size,
                              hipStream_t stream) {
    const float* x   = (const float*)d_in[0];
    const int*   ei  = (const int*)d_in[1];
    const int*   bat = (const int*)d_in[2];
    const float* Ws1 = (const float*)d_in[3];
    const float* Wd1 = (const float*)d_in[4];
    const float* as1 = (const float*)d_in[5];
    const float* ad1 = (const float*)d_in[6];
    const float* b1  = (const float*)d_in[7];
    const float* Ws2 = (const float*)d_in[8];
    const float* Wd2 = (const float*)d_in[9];
    const float* as2 = (const float*)d_in[10];
    const float* ad2 = (const float*)d_in[11];
    const float* b2  = (const float*)d_in[12];
    const float* Wl  = (const float*)d_in[13];
    const float* bl  = (const float*)d_in[14];
    float* out = (float*)d_out;

    // Workspace carve-up (~85 MB of fp32)
    float* p = (float*)d_ws;
    float* xs   = p; p += (size_t)NN * FEAT;
    float* hA   = p; p += (size_t)NN * FEAT;
    float* hB   = p; p += (size_t)NN * FEAT;
    float* aS   = p; p += (size_t)NN * HH;
    float* aD   = p; p += (size_t)NN * HH;
    unsigned* menc = (unsigned*)p; p += (size_t)NN * HH;
    float* den  = p; p += (size_t)NN * HH;
    float* ew   = p; p += (size_t)EE * HH;
    float* wsv  = p; p += (size_t)HH * FEAT;
    float* wdv  = p; p += (size_t)HH * FEAT;
    float* sums = p; p += (size_t)GG * FEAT;
    float* cnt  = p; p += (size_t)GG;

    // Layer 1: input x -> hA
    run_gat_layer(x, Ws1, Wd1, as1, ad1, b1, ei, xs, aS, aD, menc, den, ew, wsv, wdv, hA, stream);
    // Layer 2: input hA -> hB
    run_gat_layer(hA, Ws2, Wd2, as2, ad2, b2, ei, xs, aS, aD, menc, den, ew, wsv, wdv, hB, stream);

    // Mean pool + final linear
    gat_zero_kernel<<<cdiv_u((long long)GG * FEAT + GG, 256), 256, 0, stream>>>(sums, GG * FEAT + GG);
    gat_pool_kernel<<<cdiv_u((long long)NN * FEAT, 256), 256, 0, stream>>>(hB, bat, sums, cnt);
    gat_final_kernel<<<cdiv_u((long long)GG * OUTC, 64), 64, 0, stream>>>(sums, cnt, Wl, bl, out);
}